// Net_80530636800127
// MI455X (gfx1250) — compile-verified
//
#include <hip/hip_runtime.h>
#include <hip/hip_bf16.h>

typedef float v2f __attribute__((ext_vector_type(2)));
typedef float v8f __attribute__((ext_vector_type(8)));

#define F_IN 512
#define HID  32
#define NCLS 40
#define LDS_STRIDE 516   // 512 + 4 pad: conflict-free b64 A-frag reads

// ---------------- degree / norm ----------------
__global__ void deg_init_kernel(float* deg, int n) {
    int i = blockIdx.x * blockDim.x + threadIdx.x;
    if (i < n) deg[i] = 1.0f;   // self loop
}

__global__ void deg_edges_kernel(const int* __restrict__ ei, float* deg, int nedges) {
    int e = blockIdx.x * blockDim.x + threadIdx.x;
    if (e < nedges) atomicAdd(&deg[ei[nedges + e]], 1.0f);
}

__global__ void dinv_kernel(float* dinv, int n) {
    int i = blockIdx.x * blockDim.x + threadIdx.x;
    if (i < n) dinv[i] = rsqrtf(dinv[i]);   // deg >= 1 always (self loop)
}

// ---------------- GEMM1: HW[N,128] = x[N,512] @ [W1a|W1b|W2a|W2b] ----------------
__global__ void __launch_bounds__(256) gemm1_kernel(
    const float* __restrict__ x,
    const float* __restrict__ W1a, const float* __restrict__ W1b,
    const float* __restrict__ W2a, const float* __restrict__ W2b,
    float* __restrict__ HW)
{
    __shared__ float atile[16 * LDS_STRIDE];
    const int mbase = blockIdx.x * 16;

    // cooperative coalesced load: 16 contiguous rows of x (32 KB) into padded LDS
    const float* xsrc = x + (size_t)mbase * F_IN;
    for (int i = threadIdx.x; i < 16 * F_IN / 4; i += 256) {
        const float4 v = ((const float4*)xsrc)[i];
        const int idx = i * 4;
        float* p = &atile[(idx >> 9) * LDS_STRIDE + (idx & 511)];
        p[0] = v.x; p[1] = v.y; p[2] = v.z; p[3] = v.w;
    }
    __syncthreads();

    const int wave = threadIdx.x >> 5;        // 0..7 -> which 16-col output tile
    const int lane = threadIdx.x & 31;
    const float* W = (wave < 4) ? ((wave < 2) ? W1a : W1b)
                                : ((wave < 6) ? W2a : W2b);
    const int col   = ((wave & 1) << 4) + (lane & 15);  // column within this W (0..31)
    const int m     = lane & 15;
    const int khalf = (lane >> 4) << 1;                 // 0 or 2

    v8f acc = {0.f, 0.f, 0.f, 0.f, 0.f, 0.f, 0.f, 0.f};
    for (int k0 = 0; k0 < F_IN; k0 += 4) {
        const int k = k0 + khalf;
        v2f a, b;
        a.x = atile[m * LDS_STRIDE + k];
        a.y = atile[m * LDS_STRIDE + k + 1];
        b.x = W[k * HID + col];
        b.y = W[(k + 1) * HID + col];
        acc = __builtin_amdgcn_wmma_f32_16x16x4_f32(
            false, a, false, b, (short)0, acc, false, false);
    }

    // D layout: VGPR v, lane L -> row = v + 8*(L/16), col = L%16
    float* out = HW + (size_t)mbase * 128 + wave * 16 + (lane & 15);
    const int mrow = (lane >> 4) << 3;
    #pragma unroll
    for (int v = 0; v < 8; ++v)
        out[(size_t)(mrow + v) * 128] = acc[v];
}

// ---------------- aggregation 1 (128-wide) ----------------
__global__ void agg1_init_kernel(const float* __restrict__ HW,
                                 const float* __restrict__ dinv,
                                 float* __restrict__ AGG, int n) {
    int gid = blockIdx.x * blockDim.x + threadIdx.x;
    if (gid >= n * 128) return;
    const int node = gid >> 7;
    const float di = dinv[node];
    AGG[gid] = HW[gid] * di * di;   // self-loop contribution
}

__global__ void agg1_edges_kernel(const int* __restrict__ ei,
                                  const float* __restrict__ dinv,
                                  const float* __restrict__ HW,
                                  float* __restrict__ AGG, int nedges) {
    long long gid = (long long)blockIdx.x * blockDim.x + threadIdx.x;
    if (gid >= (long long)nedges * 32) return;
    const int e = (int)(gid >> 5);
    const int c = ((int)gid & 31) << 2;          // 4 columns per lane, wave covers 128
    const int s = ei[e];
    const int d = ei[nedges + e];
    const float nrm = dinv[s] * dinv[d];
    const float4 h = *(const float4*)(HW + (size_t)s * 128 + c);
    float* o = AGG + (size_t)d * 128 + c;
    atomicAdd(o + 0, h.x * nrm);
    atomicAdd(o + 1, h.y * nrm);
    atomicAdd(o + 2, h.z * nrm);
    atomicAdd(o + 3, h.w * nrm);
}

// ---------------- relu + bias + combine -> XS[N,32] ----------------
__global__ void combine_kernel(const float* __restrict__ AGG,
                               const float* __restrict__ b1a, const float* __restrict__ b1b,
                               const float* __restrict__ b2a, const float* __restrict__ b2b,
                               float* __restrict__ XS, int n) {
    int gid = blockIdx.x * blockDim.x + threadIdx.x;
    if (gid >= n * HID) return;
    const int node = gid >> 5;
    const int c    = gid & 31;
    const float* a = AGG + (size_t)node * 128;
    const float x1 = fmaxf(a[c]      + b1a[c], 0.f) + fmaxf(a[32 + c] + b1b[c], 0.f);
    const float x2 = fmaxf(a[64 + c] + b2a[c], 0.f) + fmaxf(a[96 + c] + b2b[c], 0.f);
    XS[gid] = x1 + x2;
}

// ---------------- GEMM2: Z[N,40] = XS[N,32] @ (W3a+W3b) ----------------
__global__ void __launch_bounds__(128) gemm2_kernel(
    const float* __restrict__ XS,
    const float* __restrict__ W3a, const float* __restrict__ W3b,
    float* __restrict__ Z)
{
    __shared__ float atile[16 * 36];   // 32 + 4 pad
    const int mbase = blockIdx.x * 16;
    for (int i = threadIdx.x; i < 16 * HID; i += 128)
        atile[(i >> 5) * 36 + (i & 31)] = XS[(size_t)mbase * HID + i];
    __syncthreads();

    const int wave = threadIdx.x >> 5;
    const int lane = threadIdx.x & 31;
    if (wave >= 3) return;                      // 3 waves cover cols 0..47 (40 valid)

    const int ncol  = wave * 16 + (lane & 15);
    const bool valid = ncol < NCLS;
    const int m     = lane & 15;
    const int khalf = (lane >> 4) << 1;

    v8f acc = {0.f, 0.f, 0.f, 0.f, 0.f, 0.f, 0.f, 0.f};
    for (int k0 = 0; k0 < HID; k0 += 4) {
        const int k = k0 + khalf;
        v2f a, b;
        a.x = atile[m * 36 + k];
        a.y = atile[m * 36 + k + 1];
        b.x = valid ? (W3a[k * NCLS + ncol]       + W3b[k * NCLS + ncol])       : 0.f;
        b.y = valid ? (W3a[(k + 1) * NCLS + ncol] + W3b[(k + 1) * NCLS + ncol]) : 0.f;
        acc = __builtin_amdgcn_wmma_f32_16x16x4_f32(
            false, a, false, b, (short)0, acc, false, false);
    }
    if (valid) {
        const int mrow = (lane >> 4) << 3;
        #pragma unroll
        for (int v = 0; v < 8; ++v)
            Z[(size_t)(mbase + mrow + v) * NCLS + ncol] = acc[v];
    }
}

// ---------------- aggregation 2 (40-wide) ----------------
__global__ void agg2_init_kernel(const float* __restrict__ Z,
                                 const float* __restrict__ dinv,
                                 const float* __restrict__ b3a,
                                 const float* __restrict__ b3b,
                                 float* __restrict__ AGG2, int n) {
    int gid = blockIdx.x * blockDim.x + threadIdx.x;
    if (gid >= n * NCLS) return;
    const int node = gid / NCLS;
    const int c    = gid - node * NCLS;
    const float di = dinv[node];
    AGG2[gid] = Z[gid] * di * di + b3a[c] + b3b[c];
}

__global__ void agg2_edges_kernel(const int* __restrict__ ei,
                                  const float* __restrict__ dinv,
                                  const float* __restrict__ Z,
                                  float* __restrict__ AGG2, int nedges) {
    long long gid = (long long)blockIdx.x * blockDim.x + threadIdx.x;
    if (gid >= (long long)nedges * 8) return;
    const int e     = (int)(gid >> 3);
    const int cbase = ((int)gid & 7) * 5;        // 8 threads x 5 cols = 40
    const int s = ei[e];
    const int d = ei[nedges + e];
    const float nrm = dinv[s] * dinv[d];
    const float* zs = Z + (size_t)s * NCLS + cbase;
    float* o = AGG2 + (size_t)d * NCLS + cbase;
    #pragma unroll
    for (int j = 0; j < 5; ++j)
        atomicAdd(o + j, zs[j] * nrm);
}

// ---------------- log_softmax over 40 classes ----------------
__global__ void logsoftmax_kernel(const float* __restrict__ AGG2,
                                  float* __restrict__ out, int n) {
    int i = blockIdx.x * blockDim.x + threadIdx.x;
    if (i >= n) return;
    const float* z = AGG2 + (size_t)i * NCLS;
    float m = z[0];
    #pragma unroll
    for (int c = 1; c < NCLS; ++c) m = fmaxf(m, z[c]);
    float s = 0.f;
    #pragma unroll
    for (int c = 0; c < NCLS; ++c) s += __expf(z[c] - m);
    const float ls = m + __logf(s);
    float* o = out + (size_t)i * NCLS;
    #pragma unroll
    for (int c = 0; c < NCLS; ++c) o[c] = z[c] - ls;
}

extern "C" void kernel_launch(void* const* d_in, const int* in_sizes, int n_in,
                              void* d_out, int out_size, void* d_ws, size_t ws_size,
                              hipStream_t stream) {
    const float* x   = (const float*)d_in[0];
    const int*   ei  = (const int*)  d_in[1];
    const float* W1a = (const float*)d_in[2];  const float* b1a = (const float*)d_in[3];
    const float* W1b = (const float*)d_in[4];  const float* b1b = (const float*)d_in[5];
    const float* W2a = (const float*)d_in[6];  const float* b2a = (const float*)d_in[7];
    const float* W2b = (const float*)d_in[8];  const float* b2b = (const float*)d_in[9];
    const float* W3a = (const float*)d_in[10]; const float* b3a = (const float*)d_in[11];
    const float* W3b = (const float*)d_in[12]; const float* b3b = (const float*)d_in[13];
    float* out = (float*)d_out;

    const int N = in_sizes[0] / F_IN;     // 100000
    const int E = in_sizes[1] / 2;        // 3200000

    // workspace carve-out (256B aligned)
    char* ws = (char*)d_ws;
    size_t off = 0;
    auto carve = [&](size_t bytes) {
        char* p = ws + off;
        off += (bytes + 255) & ~(size_t)255;
        return p;
    };
    float* dinv = (float*)carve((size_t)N * 4);
    float* HW   = (float*)carve((size_t)N * 128 * 4);
    float* AGG  = (float*)carve((size_t)N * 128 * 4);
    float* XS   = (float*)carve((size_t)N * HID * 4);
    float* Z    = (float*)carve((size_t)N * NCLS * 4);
    float* AGG2 = (float*)carve((size_t)N * NCLS * 4);
    (void)ws_size; (void)n_in; (void)out_size;

    const int T = 256;

    // 1) degrees -> dinv
    deg_init_kernel <<<(N + T - 1) / T, T, 0, stream>>>(dinv, N);
    deg_edges_kernel<<<(E + T - 1) / T, T, 0, stream>>>(ei, dinv, E);
    dinv_kernel     <<<(N + T - 1) / T, T, 0, stream>>>(dinv, N);

    // 2) fused 4-way projection GEMM (WMMA f32)
    gemm1_kernel<<<N / 16, 256, 0, stream>>>(x, W1a, W1b, W2a, W2b, HW);

    // 3) aggregation (self loops + edges)
    agg1_init_kernel <<<((size_t)N * 128 + T - 1) / T, T, 0, stream>>>(HW, dinv, AGG, N);
    agg1_edges_kernel<<<((size_t)E * 32 + T - 1) / T, T, 0, stream>>>(ei, dinv, HW, AGG, E);

    // 4) relu + bias + combine
    combine_kernel<<<((size_t)N * HID + T - 1) / T, T, 0, stream>>>(AGG, b1a, b1b, b2a, b2b, XS, N);

    // 5) classifier GEMM (WMMA f32), W3a+W3b fused
    gemm2_kernel<<<N / 16, 128, 0, stream>>>(XS, W3a, W3b, Z);

    // 6) aggregation 2 (self loops + bias + edges)
    agg2_init_kernel <<<((size_t)N * NCLS + T - 1) / T, T, 0, stream>>>(Z, dinv, b3a, b3b, AGG2, N);
    agg2_edges_kernel<<<((size_t)E * 8 + T - 1) / T, T, 0, stream>>>(ei, dinv, Z, AGG2, E);

    // 7) log_softmax
    logsoftmax_kernel<<<(N + T - 1) / T, T, 0, stream>>>(AGG2, out, N);
}